// Attention_9775345565701
// MI455X (gfx1250) — compile-verified
//
#include <hip/hip_runtime.h>
#include <hip/hip_bf16.h>

typedef __attribute__((ext_vector_type(2))) float v2f;
typedef __attribute__((ext_vector_type(8))) float v8f;
typedef __attribute__((ext_vector_type(4))) unsigned int u32x4;
typedef __attribute__((ext_vector_type(8))) int i32x8;
typedef __attribute__((ext_vector_type(4))) int i32x4;

// D = A(16x4) * B(4x16) + C, fp32, wave32 WMMA
#define WMMA_F32X4(a, b, c) \
  __builtin_amdgcn_wmma_f32_16x16x4_f32(false, (a), false, (b), (short)0, (c), false, false)

// LDS byte offset of a __shared__ object (flat LDS address low 32 bits = offset)
#define LDS_OFF(p) ((unsigned)(size_t)(p))

// ---------------------------------------------------------------------------
// Problem constants
// ---------------------------------------------------------------------------
#define BATCH 2
#define SEQ   2048
#define HID   2048
#define NHEAD 16
#define HDIM  128
#define QKV_LD 6144   // 3*HID

// ---------------------------------------------------------------------------
// TDM: async 2D tile load global -> LDS with optional LDS padding.
// rows x cols f32 tile, global row stride = stride_elems, tensor == tile
// (no OOB region). pad: after every 2^(pi+1) DWORDs insert (pa+1) DWORDs.
// Tracked by TENSORcnt. Issue from one wave only.
// ---------------------------------------------------------------------------
__device__ __forceinline__ void tdm_load_2d_f32(unsigned lds_off, const float* gptr,
                                                int rows, int cols, int stride_elems,
                                                unsigned pi, unsigned pa)
{
  unsigned long long ga = (unsigned long long)(size_t)gptr;
  u32x4 g0;
  g0[0] = 1u;                                    // count=1, user descriptor
  g0[1] = lds_off;                               // lds_addr
  g0[2] = (unsigned)ga;                          // global_addr[31:0]
  g0[3] = ((unsigned)(ga >> 32) & 0x1ffffffu)    // global_addr[56:32]
          | (2u << 30);                          // type = 2 ("image")
  i32x8 g1;
  g1[0] = (int)((2u << 16)                       // data_size = 4 bytes
                | (1u << 20)                     // pad_enable
                | (pi << 22)                     // pad_interval code
                | (pa << 25));                   // pad_amount code
  g1[1] = (int)(((unsigned)cols & 0xffffu) << 16);               // tensor_dim0[15:0]
  g1[2] = (int)((((unsigned)cols >> 16) & 0xffffu)               // tensor_dim0[31:16]
                | (((unsigned)rows & 0xffffu) << 16));           // tensor_dim1[15:0]
  g1[3] = (int)((((unsigned)rows >> 16) & 0xffffu)               // tensor_dim1[31:16]
                | (((unsigned)cols & 0xffffu) << 16));           // tile_dim0
  g1[4] = (int)((unsigned)rows & 0xffffu);                       // tile_dim1 (tile_dim2=0)
  g1[5] = (int)(unsigned)stride_elems;                           // tensor_dim0_stride[31:0]
  g1[6] = 0;                                                     // stride hi / dim1_stride
  g1[7] = 0;
  i32x4 z4 = {};
  i32x8 z8 = {};
  // amdgpu-toolchain (clang-23) 6-arg form
  __builtin_amdgcn_tensor_load_to_lds(g0, g1, z4, z4, z8, 0);
}

// ===========================================================================
// Kernel 1/4: tiled NT GEMM  C[M,N] = A[M,K] * B[N,K]^T   (fp32, WMMA)
// BM=128, BN=128, BK=16; 256 threads = 8 waves as 2x4; wave tile 64x32.
// Double-buffered TDM staging: wave 0 issues next tile's DMA, all waves
// compute current buffer, s_wait_tensorcnt + one barrier per stage.
// ===========================================================================
__global__ __launch_bounds__(256) void gemm_nt_f32(
    const float* __restrict__ A, const float* __restrict__ B,
    float* __restrict__ C, int M, int N, int K)
{
  const int LDT = 18; // 16 + 2 pad (TDM-inserted): conflict-free, 8B-aligned frags
  __shared__ float As[2][128 * 18];
  __shared__ float Bs[2][128 * 18];

  const int tid  = threadIdx.x;
  const int wave = tid >> 5;
  const int lane = tid & 31;
  const int lo   = lane & 15;
  const int hi   = lane >> 4;
  const int wm   = wave >> 2;   // 0..1
  const int wn   = wave & 3;    // 0..3
  const int m0   = blockIdx.y * 128;
  const int n0   = blockIdx.x * 128;

  const float* Abase = A + (size_t)m0 * K;
  const float* Bbase = B + (size_t)n0 * K;
  const unsigned aoff[2] = {LDS_OFF(&As[0][0]), LDS_OFF(&As[1][0])};
  const unsigned boff[2] = {LDS_OFF(&Bs[0][0]), LDS_OFF(&Bs[1][0])};

  v8f acc[4][2];
  for (int mt = 0; mt < 4; ++mt)
    for (int nt = 0; nt < 2; ++nt)
      acc[mt][nt] = {};

  const int nk = K >> 4;
  if (wave == 0) {
    tdm_load_2d_f32(aoff[0], Abase, 128, 16, K, 3u, 1u);
    tdm_load_2d_f32(boff[0], Bbase, 128, 16, K, 3u, 1u);
    __builtin_amdgcn_s_wait_tensorcnt(0);
  }
  __syncthreads();

  int buf = 0;
  for (int kt = 0; kt < nk; ++kt) {
    // prefetch next stage into the other buffer (reads of it finished last iter)
    if (wave == 0 && kt + 1 < nk) {
      int k0n = (kt + 1) << 4;
      tdm_load_2d_f32(aoff[buf ^ 1], Abase + k0n, 128, 16, K, 3u, 1u);
      tdm_load_2d_f32(boff[buf ^ 1], Bbase + k0n, 128, 16, K, 3u, 1u);
    }
    const float* Asb = As[buf];
    const float* Bsb = Bs[buf];
    for (int kk = 0; kk < 16; kk += 4) {
      v2f afr[4], bfr[2];
      for (int mt = 0; mt < 4; ++mt)
        afr[mt] = *(const v2f*)(Asb + (wm * 64 + mt * 16 + lo) * LDT + kk + 2 * hi);
      for (int nt = 0; nt < 2; ++nt)
        bfr[nt] = *(const v2f*)(Bsb + (wn * 32 + nt * 16 + lo) * LDT + kk + 2 * hi);
      for (int mt = 0; mt < 4; ++mt)
        for (int nt = 0; nt < 2; ++nt)
          acc[mt][nt] = WMMA_F32X4(afr[mt], bfr[nt], acc[mt][nt]);
    }
    if (wave == 0) __builtin_amdgcn_s_wait_tensorcnt(0);
    __syncthreads();
    buf ^= 1;
  }

  // epilogue: C-layout element (M = j + 8*hi, N = lo) per accumulator
  for (int mt = 0; mt < 4; ++mt) {
    int mb = m0 + wm * 64 + mt * 16;
    for (int nt = 0; nt < 2; ++nt) {
      int n = n0 + wn * 32 + nt * 16 + lo;
      for (int j = 0; j < 8; ++j)
        C[(size_t)(mb + j + 8 * hi) * N + n] = acc[mt][nt][j];
    }
  }
}

// ===========================================================================
// Kernel 2: RoPE in place on the q and k slices of qkv [B, L, 3H]
// pair (d, d+64) shares angle l * theta^(-d/64)
// ===========================================================================
__global__ __launch_bounds__(256) void rope_kernel(float* __restrict__ qkv)
{
  int idx = blockIdx.x * 256 + threadIdx.x;   // B*L*NH*64 threads
  int d = idx & 63;
  int h = (idx >> 6) & (NHEAD - 1);
  int l = (idx >> 10) & (SEQ - 1);
  int b = idx >> 21;

  float inv_freq = __powf(10000.0f, -(float)d * (1.0f / 64.0f));
  float s, c;
  __sincosf((float)l * inv_freq, &s, &c);

  float* q = qkv + ((size_t)(b * SEQ + l)) * QKV_LD + h * HDIM;
  float* k = q + HID;
  float q1 = q[d], q2 = q[d + 64];
  q[d]      = q1 * c - q2 * s;
  q[d + 64] = q2 * c + q1 * s;
  float k1 = k[d], k2 = k[d + 64];
  k[d]      = k1 * c - k2 * s;
  k[d + 64] = k2 * c + k1 * s;
}

// ===========================================================================
// Kernel 3: causal flash attention, fp32 WMMA.
// grid = (L/128, NH, B), 256 threads = 8 waves, wave owns 16 query rows.
// V tile staged by TDM (overlapped with manual transposed K staging);
// Q frags live in registers (pre-scaled by 1/sqrt(HD)).
// Output written as [B, L, NH*HD] so the final GEMM is a plain NT GEMM.
// ===========================================================================
__global__ __launch_bounds__(256) void flash_attn_f32(
    const float* __restrict__ qkv, float* __restrict__ out)
{
  __shared__ float Kt[128 * 33];        // K transposed: [d][kv], pad 33
  __shared__ float Vs[32 * 132];        // V natural:    [kv][d], pad 132 (TDM pad)
  __shared__ float Ps[8 * 16 * 34];     // per-wave P:   [16][34]

  const int tid  = threadIdx.x;
  const int wave = tid >> 5;
  const int lane = tid & 31;
  const int lo   = lane & 15;
  const int hi   = lane >> 4;

  const int b  = blockIdx.z;
  const int h  = blockIdx.y;
  const int q0 = blockIdx.x * 128;
  const int qr = q0 + wave * 16;        // this wave's first query row

  const float scale = 0.08838834764831843f; // 1/sqrt(128)

  const float* Qbase = qkv + ((size_t)(b * SEQ)) * QKV_LD + h * HDIM;
  const float* Kbase = Qbase + HID;
  const float* Vbase = Qbase + 2 * HID;
  const unsigned vs_off = LDS_OFF(&Vs[0]);

  // Q A-fragments, pre-scaled: aq[dt] holds Q[qr+lo][dt*4 + 2*hi + {0,1}]
  v2f aq[32];
  {
    const float* qrow = Qbase + (size_t)(qr + lo) * QKV_LD;
    for (int dt = 0; dt < 32; ++dt) {
      v2f t = *(const v2f*)(qrow + dt * 4 + 2 * hi);
      aq[dt] = t * scale;
    }
  }

  v8f o_acc[8];
  for (int nt = 0; nt < 8; ++nt) o_acc[nt] = {};
  float m_st[8], l_st[8];
  for (int j = 0; j < 8; ++j) { m_st[j] = -1e30f; l_st[j] = 0.0f; }

  float* Pw = Ps + wave * 16 * 34;

  const int nblk = (q0 + 128) >> 5;     // causal upper bound for this block
  for (int t = 0; t < nblk; ++t) {
    const int kv0 = t * 32;
    __syncthreads();                    // all reads of previous K/V done
    // async: V tile 32x128 -> Vs with pad to stride 132 (interval 128 dw, pad 4 dw)
    if (wave == 0)
      tdm_load_2d_f32(vs_off, Vbase + (size_t)kv0 * QKV_LD, 32, 128, QKV_LD, 6u, 3u);
    // meanwhile: stage K transposed (32x128 floats, 4 float4/thread)
    for (int i = 0; i < 4; ++i) {
      int f = tid + i * 256;            // 0..1023
      int r = f >> 5;                   // kv row 0..31
      int c = (f & 31) << 2;            // d col, multiple of 4
      float4 kvv = *(const float4*)(Kbase + (size_t)(kv0 + r) * QKV_LD + c);
      Kt[(c + 0) * 33 + r] = kvv.x;
      Kt[(c + 1) * 33 + r] = kvv.y;
      Kt[(c + 2) * 33 + r] = kvv.z;
      Kt[(c + 3) * 33 + r] = kvv.w;
    }
    if (wave == 0) __builtin_amdgcn_s_wait_tensorcnt(0);
    __syncthreads();

    if (kv0 <= qr + 15) {               // wave has unmasked work in this block
      // S = (Q*scale) @ K^T : 16x32 tile as two 16x16 accumulators
      v8f s_acc[2];
      s_acc[0] = {}; s_acc[1] = {};
      for (int dt = 0; dt < 32; ++dt) {
        int krow = dt * 4 + 2 * hi;
        for (int nt = 0; nt < 2; ++nt) {
          v2f bf;
          bf.x = Kt[(krow + 0) * 33 + nt * 16 + lo];
          bf.y = Kt[(krow + 1) * 33 + nt * 16 + lo];
          s_acc[nt] = WMMA_F32X4(aq[dt], bf, s_acc[nt]);
        }
      }
      // causal mask on the diagonal block
      if (kv0 + 31 > qr) {
        for (int nt = 0; nt < 2; ++nt) {
          int n = kv0 + nt * 16 + lo;
          for (int j = 0; j < 8; ++j) {
            int m = qr + j + 8 * hi;
            if (n > m) s_acc[nt][j] = -1e30f;
          }
        }
      }
      // online softmax (rows live across 16 lanes of each half-wave)
      float alpha[8];
      for (int j = 0; j < 8; ++j) {
        float v = fmaxf(s_acc[0][j], s_acc[1][j]);
        for (int off = 1; off < 16; off <<= 1) v = fmaxf(v, __shfl_xor(v, off, 32));
        float m_new = fmaxf(m_st[j], v);
        alpha[j] = __expf(m_st[j] - m_new);
        float p0 = __expf(s_acc[0][j] - m_new);
        float p1 = __expf(s_acc[1][j] - m_new);
        s_acc[0][j] = p0; s_acc[1][j] = p1;
        float rs = p0 + p1;
        for (int off = 1; off < 16; off <<= 1) rs += __shfl_xor(rs, off, 32);
        l_st[j] = l_st[j] * alpha[j] + rs;
        m_st[j] = m_new;
      }
      for (int nt = 0; nt < 8; ++nt)
        for (int j = 0; j < 8; ++j)
          o_acc[nt][j] *= alpha[j];
      // P: C-layout -> A-layout via per-wave LDS
      for (int nt = 0; nt < 2; ++nt)
        for (int j = 0; j < 8; ++j)
          Pw[(j + 8 * hi) * 34 + nt * 16 + lo] = s_acc[nt][j];
      asm volatile("s_wait_dscnt 0" ::: "memory");
      // O += P @ V
      for (int kt = 0; kt < 8; ++kt) {
        v2f pa = *(const v2f*)(Pw + lo * 34 + kt * 4 + 2 * hi);
        int vr = kt * 4 + 2 * hi;
        for (int nt = 0; nt < 8; ++nt) {
          v2f vb;
          vb.x = Vs[(vr + 0) * 132 + nt * 16 + lo];
          vb.y = Vs[(vr + 1) * 132 + nt * 16 + lo];
          o_acc[nt] = WMMA_F32X4(pa, vb, o_acc[nt]);
        }
      }
    }
  }

  // epilogue: O /= l, write [B, L, NH*HD]
  for (int j = 0; j < 8; ++j) {
    float inv_l = 1.0f / l_st[j];
    int m = qr + j + 8 * hi;
    float* orow = out + ((size_t)(b * SEQ + m)) * HID + h * HDIM;
    for (int nt = 0; nt < 8; ++nt)
      orow[nt * 16 + lo] = o_acc[nt][j] * inv_l;
  }
}

// ===========================================================================
// Host launch
// ===========================================================================
extern "C" void kernel_launch(void* const* d_in, const int* in_sizes, int n_in,
                              void* d_out, int out_size, void* d_ws, size_t ws_size,
                              hipStream_t stream) {
  (void)in_sizes; (void)n_in; (void)out_size; (void)ws_size;

  const float* x     = (const float*)d_in[0];   // [B, L, H]
  const float* w_qkv = (const float*)d_in[1];   // [3H, H]
  const float* w_o   = (const float*)d_in[2];   // [H, H]
  float* out = (float*)d_out;                   // [B, L, H]

  float* qkv  = (float*)d_ws;                         // [B*L, 3H] = 100.7 MB
  float* attn = qkv + (size_t)(BATCH * SEQ) * QKV_LD; // [B*L, H]  =  33.6 MB

  const int M = BATCH * SEQ;  // 4096

  // 1) qkv = x @ w_qkv^T
  gemm_nt_f32<<<dim3(QKV_LD / 128, M / 128), 256, 0, stream>>>(
      x, w_qkv, qkv, M, QKV_LD, HID);

  // 2) RoPE in place on q,k slices
  rope_kernel<<<(BATCH * SEQ * NHEAD * 64) / 256, 256, 0, stream>>>(qkv);

  // 3) causal flash attention -> attn [B*L, H]
  flash_attn_f32<<<dim3(SEQ / 128, NHEAD, BATCH), 256, 0, stream>>>(qkv, attn);

  // 4) out = attn @ w_o^T
  gemm_nt_f32<<<dim3(HID / 128, M / 128), 256, 0, stream>>>(
      attn, w_o, out, M, HID, HID);
}